// classic_slab1D_eqx_difx_64948495450701
// MI455X (gfx1250) — compile-verified
//
#include <hip/hip_runtime.h>

// ---------------------------------------------------------------------------
// Slab model:  z_{n+1} = c*z_n + f_n  (complex), z = U + iV,
//   c  = (1 - dt*K1) - i*(dt*fc),   f_n = dt*K0*(tax_n + i*tay_n)
// Parallelized as a hierarchical affine-map scan; inner 16-step tiles solved
// as 16x16 triangular-Toeplitz matmuls on the WMMA pipe (V_WMMA_F32_16X16X4).
// Output is 80 MB f32 -> store-bandwidth bound (~3.4us @ 23.3 TB/s).
// ---------------------------------------------------------------------------

typedef __attribute__((ext_vector_type(2))) float v2f;
typedef __attribute__((ext_vector_type(4))) float v4f;
typedef __attribute__((ext_vector_type(8))) float v8f;

#define NSUB 60              // forcing substeps per sample (3600/60)
#define TILE 16              // steps per tile (WMMA M dimension)
#define TPB_TILES 128        // tiles per block (8 waves * 16 columns)
#define SEG 20               // block-offsets per thread in the scan kernel

struct cplx { float re, im; };
__device__ __forceinline__ cplx cmul(cplx a, cplx b) {
    return { a.re * b.re - a.im * b.im, a.re * b.im + a.im * b.re };
}
__device__ __forceinline__ cplx cadd(cplx a, cplx b) { return { a.re + b.re, a.im + b.im }; }
// m*x + a
__device__ __forceinline__ cplx cfma(cplx m, cplx x, cplx a) {
    return { fmaf(m.re, x.re, fmaf(-m.im, x.im, a.re)),
             fmaf(m.re, x.im, fmaf( m.im, x.re, a.im)) };
}

// forcing f_n = dt*K0 * (tax_n, tay_n), linear interp between samples
__device__ __forceinline__ cplx forcing(int n, int nsteps, int nl,
                                        const float* __restrict__ TAx,
                                        const float* __restrict__ TAy,
                                        float dtK0) {
    if (n > nsteps - 1) n = nsteps - 1;           // clamp for padded tiles
    int itf = n / NSUB;
    float aa = (float)(n - itf * NSUB) * (1.0f / (float)NSUB);
    int i1 = itf + 1; if (i1 > nl - 1) i1 = nl - 1;   // itsup=-1 == last sample
    if (itf > nl - 1) itf = nl - 1;
    float x0 = TAx[itf], y0 = TAy[itf];
    float tax = fmaf(aa, TAx[i1] - x0, x0);
    float tay = fmaf(aa, TAy[i1] - y0, y0);
    return { dtK0 * tax, dtK0 * tay };
}

// Header in d_ws: [0]=c.re [1]=c.im [2]=dt*K0 [3]=pad, [4+2j],[5+2j]=c^(16*2^j), j=0..7
__global__ void k_setup(const float* __restrict__ pk, const float* __restrict__ fcp,
                        float* __restrict__ hdr) {
    if (threadIdx.x == 0 && blockIdx.x == 0) {
        double K0 = exp((double)pk[0]);
        double K1 = exp((double)pk[1]);
        double dt = 60.0;
        double cr = 1.0 - dt * K1;
        double ci = -dt * (double)fcp[0];
        hdr[0] = (float)cr; hdr[1] = (float)ci; hdr[2] = (float)(dt * K0); hdr[3] = 0.0f;
        double pr = cr, pi = ci;
        for (int s = 0; s < 4; ++s) { double nr = pr*pr - pi*pi, ni = 2.0*pr*pi; pr = nr; pi = ni; } // c^16
        for (int j = 0; j < 8; ++j) {
            hdr[4 + 2*j] = (float)pr; hdr[5 + 2*j] = (float)pi;
            double nr = pr*pr - pi*pi, ni = 2.0*pr*pi; pr = nr; pi = ni;
        }
    }
}

// Pass 1: per-block affine offset (over 128 tiles = 2048 steps)
__global__ void __launch_bounds__(TPB_TILES)
k_blocktot(const float* __restrict__ TAx, const float* __restrict__ TAy,
           const float* __restrict__ hdr, float2* __restrict__ blockTot,
           int nsteps, int nt, int nl) {
    __shared__ float2 red[TPB_TILES];
    cplx c = { hdr[0], hdr[1] };
    float dtK0 = hdr[2];
    int t = blockIdx.x * TPB_TILES + threadIdx.x;
    cplx w = { 0.0f, 0.0f };
    if (t < nt) {
        int n0 = t * TILE;
#pragma unroll
        for (int j = 0; j < TILE; ++j) w = cfma(c, w, forcing(n0 + j, nsteps, nl, TAx, TAy, dtK0));
    }
    red[threadIdx.x] = make_float2(w.re, w.im);
    __syncthreads();
    int jlev = 0;
    for (int s = 1; s < TPB_TILES; s <<= 1, ++jlev) {
        cplx W = { hdr[4 + 2*jlev], hdr[5 + 2*jlev] };   // c^(16*s)
        if ((threadIdx.x & (2*s - 1)) == 0) {
            cplx lo = { red[threadIdx.x].x, red[threadIdx.x].y };
            cplx hi = { red[threadIdx.x + s].x, red[threadIdx.x + s].y };
            cplx r = cfma(W, lo, hi);                    // later * earlier + later-offset
            red[threadIdx.x] = make_float2(r.re, r.im);
        }
        __syncthreads();
    }
    if (threadIdx.x == 0) blockTot[blockIdx.x] = red[0];
}

// Pass 2: scan block offsets -> per-block start states (one block)
__global__ void __launch_bounds__(256)
k_scan(const float* __restrict__ hdr, const float2* __restrict__ blockTot,
       float2* __restrict__ sBlock, int nblk) {
    __shared__ float2 lds[256];
    int tid = threadIdx.x;
    cplx Cb = { hdr[4 + 2*7], hdr[5 + 2*7] };            // c^2048 (per-block multiplier)
    cplx Cseg = { 1.0f, 0.0f }; cplx base = Cb;
    int e = SEG;
    while (e) { if (e & 1) Cseg = cmul(Cseg, base); base = cmul(base, base); e >>= 1; }
    int k0 = tid * SEG;
    int k1 = k0 + SEG; if (k1 > nblk) k1 = nblk;
    cplx o = { 0.0f, 0.0f };
    for (int k = k0; k < k1; ++k) { float2 v = blockTot[k]; o = cfma(Cb, o, cplx{ v.x, v.y }); }
    lds[tid] = make_float2(o.re, o.im);
    __syncthreads();
    cplx W = Cseg;
    for (int s = 1; s < 256; s <<= 1) {
        cplx v = { lds[tid].x, lds[tid].y };
        cplx r = v;
        if (tid >= s) { cplx lo = { lds[tid - s].x, lds[tid - s].y }; r = cfma(W, lo, v); }
        __syncthreads();
        lds[tid] = make_float2(r.re, r.im);
        __syncthreads();
        W = cmul(W, W);
    }
    cplx S = (tid == 0) ? cplx{ 0.0f, 0.0f } : cplx{ lds[tid - 1].x, lds[tid - 1].y };
    for (int k = k0; k < k1; ++k) {
        sBlock[k] = make_float2(S.re, S.im);
        float2 v = blockTot[k];
        S = cfma(Cb, S, cplx{ v.x, v.y });
    }
}

// Pass 3: per-tile starts (LDS scan) + WMMA triangular-Toeplitz solve + store
__global__ void __launch_bounds__(256)
k_solve(const float* __restrict__ TAx, const float* __restrict__ TAy,
        const float* __restrict__ hdr, const float2* __restrict__ sBlock,
        float* __restrict__ Us, float* __restrict__ Vs,
        int nsteps, int nt, int nl) {
    __shared__ float2 wloc[TPB_TILES];
    __shared__ float2 tstart[TPB_TILES];
    int tid = threadIdx.x;
    cplx c = { hdr[0], hdr[1] };
    float dtK0 = hdr[2];
    int tbase = blockIdx.x * TPB_TILES;

    // Phase A: per-tile local offsets
    if (tid < TPB_TILES) {
        int t = tbase + tid;
        cplx w = { 0.0f, 0.0f };
        if (t < nt) {
            int n0 = t * TILE;
#pragma unroll
            for (int j = 0; j < TILE; ++j) w = cfma(c, w, forcing(n0 + j, nsteps, nl, TAx, TAy, dtK0));
        }
        wloc[tid] = make_float2(w.re, w.im);
    }
    __syncthreads();

    // Phase B: inclusive Kogge-Stone scan, weight c^(16*s)
    int jlev = 0;
    for (int s = 1; s < TPB_TILES; s <<= 1, ++jlev) {
        float2 r = make_float2(0.0f, 0.0f);
        if (tid < TPB_TILES) {
            cplx W = { hdr[4 + 2*jlev], hdr[5 + 2*jlev] };
            cplx v = { wloc[tid].x, wloc[tid].y };
            cplx rr = v;
            if (tid >= s) { cplx lo = { wloc[tid - s].x, wloc[tid - s].y }; rr = cfma(W, lo, v); }
            r = make_float2(rr.re, rr.im);
        }
        __syncthreads();
        if (tid < TPB_TILES) wloc[tid] = r;
        __syncthreads();
    }
    // per-tile start state: c^(16*tid) * S_block + excl_scan
    if (tid < TPB_TILES) {
        cplx m = { 1.0f, 0.0f };
#pragma unroll
        for (int j = 0; j < 7; ++j)
            if (tid & (1 << j)) m = cmul(m, cplx{ hdr[4 + 2*j], hdr[5 + 2*j] });
        float2 sb = sBlock[blockIdx.x];
        cplx st = cmul(m, cplx{ sb.x, sb.y });
        if (tid > 0) st = cadd(st, cplx{ wloc[tid - 1].x, wloc[tid - 1].y });
        tstart[tid] = make_float2(st.re, st.im);
    }
    __syncthreads();

    // Phase C: one wave solves 16 tiles (256 steps) with 16 WMMAs.
    int wvid = tid >> 5, lane = tid & 31;
    int half = lane >> 4, m16 = lane & 15;   // m16 = A row M == B/C/D column N
    int tileL = wvid * 16 + m16;
    int tile  = tbase + tileL;

    // A = lower-triangular Toeplitz of c powers: T[i][j] = c^(i-j), i>=j.
    // Lane layout (16x16x4 A): row M=m16; K = 4*kk + 2*half + {0,1}.
    int d0 = m16 - 2 * half;
    cplx pw[16];
    {
        cplx cur = { 1.0f, 0.0f };
#pragma unroll
        for (int k = 15; k >= 0; --k) {       // ascending exponent d = d0-k
            int d = d0 - k;
            if (d >= 0) { pw[k] = cur; cur = cmul(cur, c); }
            else        { pw[k] = cplx{ 0.0f, 0.0f }; }
        }
    }
    v2f Ar[4], Ai[4], Ain[4];
#pragma unroll
    for (int kk = 0; kk < 4; ++kk) {
        Ar[kk]  = (v2f){  pw[4*kk].re,  pw[4*kk + 1].re };
        Ai[kk]  = (v2f){  pw[4*kk].im,  pw[4*kk + 1].im };
        Ain[kk] = (v2f){ -pw[4*kk].im, -pw[4*kk + 1].im };
    }

    // B columns = forcing of this lane's tile; rows K = 4*kk + 2*half + {0,1}
    v2f Br[4], Bi[4];
    int nbase = tile * TILE + 2 * half;
#pragma unroll
    for (int kk = 0; kk < 4; ++kk) {
        cplx f0 = forcing(nbase + 4*kk,     nsteps, nl, TAx, TAy, dtK0);
        cplx f1 = forcing(nbase + 4*kk + 1, nsteps, nl, TAx, TAy, dtK0);
        Br[kk] = (v2f){ f0.re, f1.re };
        Bi[kk] = (v2f){ f0.im, f1.im };
    }

    // C seed: C[i][col] = c^(i+1) * z_start(col); lane holds rows i = r + 8*half
    float2 z0f = tstart[tileL];
    cplx z = { z0f.x, z0f.y };
    cplx cc2 = cmul(c, c), cc4 = cmul(cc2, cc2), cc8 = cmul(cc4, cc4);
    cplx mul = half ? cmul(cc8, c) : c;      // c^9 or c^1
    cplx wacc = cmul(mul, z);
    v8f accRe, accIm;
#pragma unroll
    for (int r = 0; r < 8; ++r) {
        accRe[r] = wacc.re; accIm[r] = wacc.im;
        wacc = cmul(c, wacc);
    }

    // Complex GEMM via 4 real products, K=16 chained as 4x WMMA 16x16x4 each:
    //   Re(Z) = Tr*Re(g) - Ti*Im(g) + Re(seed);  Im(Z) = Ti*Re(g) + Tr*Im(g) + Im(seed)
#pragma unroll
    for (int kk = 0; kk < 4; ++kk) {
        accRe = __builtin_amdgcn_wmma_f32_16x16x4_f32(false, Ar[kk],  false, Br[kk], (short)0, accRe, false, false);
        accRe = __builtin_amdgcn_wmma_f32_16x16x4_f32(false, Ain[kk], false, Bi[kk], (short)0, accRe, false, false);
        accIm = __builtin_amdgcn_wmma_f32_16x16x4_f32(false, Ai[kk],  false, Br[kk], (short)0, accIm, false, false);
        accIm = __builtin_amdgcn_wmma_f32_16x16x4_f32(false, Ar[kk],  false, Bi[kk], (short)0, accIm, false, false);
    }

    // Store: lane holds D rows r+8*half of its column -> 8 consecutive outputs.
    if (tile < nt) {
        int ob = tile * TILE + 8 * half;
        v4f r0 = { accRe[0], accRe[1], accRe[2], accRe[3] };
        v4f r1 = { accRe[4], accRe[5], accRe[6], accRe[7] };
        v4f i0 = { accIm[0], accIm[1], accIm[2], accIm[3] };
        v4f i1 = { accIm[4], accIm[5], accIm[6], accIm[7] };
        *(v4f*)(Us + ob)     = r0;
        *(v4f*)(Us + ob + 4) = r1;
        *(v4f*)(Vs + ob)     = i0;
        *(v4f*)(Vs + ob + 4) = i1;
    }
}

extern "C" void kernel_launch(void* const* d_in, const int* in_sizes, int n_in,
                              void* d_out, int out_size, void* d_ws, size_t ws_size,
                              hipStream_t stream) {
    const float* pk  = (const float*)d_in[0];
    const float* TAx = (const float*)d_in[1];
    const float* TAy = (const float*)d_in[2];
    const float* fcp = (const float*)d_in[3];
    int nl     = in_sizes[1];
    int nsteps = out_size / 2;           // (2, nsteps) stacked output
    int nt     = nsteps / TILE;          // 625,000 tiles (10M steps)
    int nblk   = (nt + TPB_TILES - 1) / TPB_TILES;

    float*  hdr      = (float*)d_ws;                         // 80 B header (256 B reserved)
    float2* blockTot = (float2*)((char*)d_ws + 256);
    float2* sBlock   = blockTot + nblk;                      // total ws ~80 KB

    float* Us = (float*)d_out;
    float* Vs = Us + nsteps;

    k_setup   <<<1, 32, 0, stream>>>(pk, fcp, hdr);
    k_blocktot<<<nblk, TPB_TILES, 0, stream>>>(TAx, TAy, hdr, blockTot, nsteps, nt, nl);
    k_scan    <<<1, 256, 0, stream>>>(hdr, blockTot, sBlock, nblk);
    k_solve   <<<nblk, 256, 0, stream>>>(TAx, TAy, hdr, sBlock, Us, Vs, nsteps, nt, nl);
}